// VelocityDerivatives_12223476924444
// MI455X (gfx1250) — compile-verified
//
#include <hip/hip_runtime.h>

// Vortex-particle velocity + 1st/2nd derivative features (12 per point).
// ALU/transcendental bound: per (point,particle) pair -> 1 v_rsq_f32,
// 1 v_exp_f32, ~30 FMAs. Particle table (64 x float4 = 1KB) is staged
// into LDS via the CDNA5 async global->LDS DMA path (ASYNCcnt).

#define KNPART 64

typedef __attribute__((ext_vector_type(2))) float v2f;
typedef __attribute__((ext_vector_type(4))) float v4f;

__global__ __launch_bounds__(256)
void vortex_features_kernel(const float* __restrict__ inp,   // [B, 64, 4] (y,x,tau,sig)
                            const float* __restrict__ pts,   // [B, N, 2]  (py,px)
                            float* __restrict__ out,         // [B, N, 12]
                            int npts)
{
    __shared__ v4f part[KNPART];   // after prep: (y, x, tau, 1/sig^2)

    const int b   = blockIdx.y;
    const int tid = threadIdx.x;
    const int idx = blockIdx.x * blockDim.x + tid;

    // --- Stage particle table into LDS with async DMA (gfx1250 path). ---
    if (tid < KNPART) {
        const v4f* src = (const v4f*)(inp + (size_t)b * KNPART * 4) + tid;
        unsigned lds_off = (unsigned)(size_t)(&part[tid]);  // low 32 bits = LDS byte offset
        asm volatile("global_load_async_to_lds_b128 %0, %1, off"
                     :: "v"(lds_off), "v"(src) : "memory");
    }

    // Overlap: pull this thread's point while the async copy is in flight.
    const int pidx = idx < npts ? idx : npts - 1;
    v2f p = __builtin_nontemporal_load((const v2f*)(pts + (size_t)b * npts * 2) + pidx);
    const float py = p.x, px = p.y;

    asm volatile("s_wait_asynccnt 0x0" ::: "memory");
    __syncthreads();

    // Pre-fold sig -> 1/sig^2 so the inner loop has no division/reciprocal.
    if (tid < KNPART) {
        v4f q = part[tid];
        q.w = 1.0f / (q.w * q.w);
        part[tid] = q;
    }
    __syncthreads();

    // 12 running sums (the other 3 outputs are exact negatives).
    float S_sdx = 0.f, S_sdy = 0.f, S_s = 0.f;
    float S_adydx = 0.f, S_adxdx = 0.f, S_adydy = 0.f, S_adx = 0.f, S_ady = 0.f;
    float S_byyx = 0.f, S_bxxx = 0.f, S_byxx = 0.f, S_byyy = 0.f;

    const float NLOG2E = -1.4426950408889634f;

#pragma unroll 4
    for (int i = 0; i < KNPART; ++i) {
        v4f q = part[i];                       // broadcast ds_load_b128
        float dy  = py - q.x;
        float dx  = px - q.y;
        float dy2 = dy * dy;
        float r2  = __builtin_fmaf(dx, dx, dy2);
        float rinv = __builtin_amdgcn_rsqf(r2);        // v_rsq_f32
        float ir2  = rinv * rinv;
        float is2  = q.w;
        // exp(-r2/sig^2) = exp2(-r2*is2*log2 e)
        float E = __builtin_amdgcn_exp2f(r2 * is2 * NLOG2E);   // v_exp_f32
        float s = q.z * E * rinv;              // strength
        float h = 0.5f * ir2;
        float c = -(is2 + h);                  // ds/dr2 = s*c
        float a = s * c;
        float bb = s * __builtin_fmaf(c, c, h * ir2);  // da/dr2

        float ady = a * dy, adx = a * dx;
        float bdy = bb * dy, bdx = bb * dx;
        float byy = bdy * dy, bxx = bdx * dx, byx = bdy * dx;

        S_sdx   = __builtin_fmaf(s,   dx, S_sdx);
        S_sdy   = __builtin_fmaf(s,   dy, S_sdy);
        S_s    += s;
        S_adydx = __builtin_fmaf(ady, dx, S_adydx);
        S_adxdx = __builtin_fmaf(adx, dx, S_adxdx);
        S_adydy = __builtin_fmaf(ady, dy, S_adydy);
        S_adx  += adx;
        S_ady  += ady;
        S_byyx  = __builtin_fmaf(byy, dx, S_byyx);
        S_bxxx  = __builtin_fmaf(bxx, dx, S_bxxx);
        S_byxx  = __builtin_fmaf(byx, dx, S_byxx);
        S_byyy  = __builtin_fmaf(byy, dy, S_byyy);
    }

    // Assemble the 12 features (torch/jax stacking order).
    float v0   = S_sdx;                                   // v
    float u0   = -S_sdy;                                  // u
    float J00  = 2.0f * S_adydx;                          // dv/dy
    float J01  = __builtin_fmaf(2.0f, S_adxdx, S_s);      // dv/dx
    float J10  = -__builtin_fmaf(2.0f, S_adydy, S_s);     // du/dy
    float J11  = -J00;                                    // du/dx
    float H000 = __builtin_fmaf(4.0f, S_byyx, 2.0f * S_adx);  // d2v/dy2
    float H011 = __builtin_fmaf(4.0f, S_bxxx, 6.0f * S_adx);  // d2v/dx2
    float H001 = __builtin_fmaf(4.0f, S_byxx, 2.0f * S_ady);  // d2v/dxdy
    float H100 = -__builtin_fmaf(4.0f, S_byyy, 6.0f * S_ady); // d2u/dy2
    float H111 = -H001;                                   // d2u/dx2
    float H110 = -H000;                                   // d2u/dydx

    if (idx < npts) {
        v4f* o = (v4f*)(out + ((size_t)b * npts + (size_t)idx) * 12);
        v4f o0 = { v0,   u0,   J00,  J01  };
        v4f o1 = { J10,  J11,  H000, H011 };
        v4f o2 = { H001, H100, H111, H110 };
        __builtin_nontemporal_store(o0, o + 0);
        __builtin_nontemporal_store(o1, o + 1);
        __builtin_nontemporal_store(o2, o + 2);
    }
}

extern "C" void kernel_launch(void* const* d_in, const int* in_sizes, int n_in,
                              void* d_out, int out_size, void* d_ws, size_t ws_size,
                              hipStream_t stream) {
    const float* inp = (const float*)d_in[0];   // [B,64,4]
    const float* pts = (const float*)d_in[1];   // [B,N,2]
    float* out = (float*)d_out;                 // [B,N,12]

    int batch = in_sizes[0] / (KNPART * 4);
    int npts  = in_sizes[1] / (2 * batch);

    dim3 block(256);
    dim3 grid((npts + 255) / 256, batch);
    vortex_features_kernel<<<grid, block, 0, stream>>>(inp, pts, out, npts);
}